// GCNModel_17635135718109
// MI455X (gfx1250) — compile-verified
//
#include <hip/hip_runtime.h>
#include <hip/hip_bf16.h>

typedef float v2f __attribute__((ext_vector_type(2)));
typedef float v8f __attribute__((ext_vector_type(8)));

#define BN_EPS 1e-5f

// ---------------- utility fills ----------------
__global__ void fill_f32(float* p, long long n, float v) {
    long long i = blockIdx.x * (long long)blockDim.x + threadIdx.x;
    if (i < n) p[i] = v;
}

// ---------------- graph norm ----------------
__global__ void degree_kernel(const int* __restrict__ dst, float* __restrict__ deg, int E) {
    int e = blockIdx.x * blockDim.x + threadIdx.x;
    if (e < E) atomicAdd(&deg[dst[e]], 1.0f);
}

__global__ void dinv_kernel(const float* __restrict__ deg, float* __restrict__ dinv, int n) {
    int i = blockIdx.x * blockDim.x + threadIdx.x;
    if (i < n) {
        float d = deg[i];
        dinv[i] = (d > 0.0f) ? rsqrtf(d) : 0.0f;
    }
}

__global__ void norm_kernel(const int* __restrict__ src, const int* __restrict__ dst,
                            const float* __restrict__ dinv, float* __restrict__ nrm, int E) {
    int e = blockIdx.x * blockDim.x + threadIdx.x;
    if (e < E) nrm[e] = dinv[src[e]] * dinv[dst[e]];
}

// ---------------- FP32 WMMA GEMM: C[MxNc] = A[MxK] @ B[KxNc] (+bias)(+relu) ----------------
// One wave32 per 16x16 output tile. M must be a multiple of 16; K multiple of 4; Nc multiple of 16.
// A-frag (16x4): lane m=lane&15; vgpr0 = K(k+2*half), vgpr1 = K(k+2*half+1)  [half = lane>>4]
// B-frag (4x16): lane n=lane&15; vgpr0 = row k+2*half, vgpr1 = row k+2*half+1
// D (16x16 f32): vgpr v holds row (v + 8*half), col n=lane&15
__global__ void wmma_gemm_f32(const float* __restrict__ A, const float* __restrict__ B,
                              float* __restrict__ C, const float* __restrict__ bias,
                              int M, int K, int Nc, int relu) {
    int wave = blockIdx.x * (blockDim.x >> 5) + (threadIdx.x >> 5);
    int lane = threadIdx.x & 31;
    int tilesN = Nc >> 4;
    int tileM = wave / tilesN;
    int tileN = wave - tileM * tilesN;
    if (tileM * 16 >= M) return;                 // uniform per wave -> EXEC stays all-1s
    int half = lane >> 4;
    int m = tileM * 16 + (lane & 15);
    int n = tileN * 16 + (lane & 15);

    v8f acc = {};
    const float* arow = A + (long long)m * K;
    for (int k = 0; k < K; k += 4) {
        v2f a, b;
        a.x = arow[k + 2 * half];
        a.y = arow[k + 2 * half + 1];
        b.x = B[(long long)(k + 2 * half) * Nc + n];
        b.y = B[(long long)(k + 2 * half + 1) * Nc + n];
        acc = __builtin_amdgcn_wmma_f32_16x16x4_f32(
            /*neg_a=*/false, a, /*neg_b=*/false, b,
            /*c_mod=*/(short)0, acc, /*reuse_a=*/false, /*reuse_b=*/false);
    }

    float bv = bias ? bias[n] : 0.0f;
    int rowbase = tileM * 16 + 8 * half;
#pragma unroll
    for (int v = 0; v < 8; ++v) {
        float val = acc[v] + bv;
        if (relu) val = val > 0.0f ? val : 0.0f;
        C[(long long)(rowbase + v) * Nc + n] = val;
    }
}

// ---------------- self-loop init: AGG = dinv^2 * H (full store, no pre-zero needed) -------
__global__ void selfloop_init(const float* __restrict__ H, const float* __restrict__ dinv,
                              float* __restrict__ AGG, int n, int d) {
    long long t = blockIdx.x * (long long)blockDim.x + threadIdx.x;
    if (t >= (long long)n * d) return;
    int i = (int)(t / d);
    float di = dinv[i];
    AGG[t] = di * di * H[t];
}

// ---------------- edge scatter: AGG[dst] += norm * H[src] (float4 + f32 atomics) ----------
__global__ void edge_scatter(const int* __restrict__ src, const int* __restrict__ dst,
                             const float* __restrict__ nrm, const float* __restrict__ H,
                             float* __restrict__ AGG, int E, int d) {
    int vecs = d >> 2;
    long long t = blockIdx.x * (long long)blockDim.x + threadIdx.x;
    long long total = (long long)E * vecs;
    if (t >= total) return;
    int e = (int)(t / vecs);
    int f4 = (int)(t - (long long)e * vecs) << 2;
    float w = nrm[e];
    const float4 v = *(const float4*)(H + (long long)src[e] * d + f4);
    float* out = AGG + (long long)dst[e] * d + f4;
    atomicAdd(out + 0, w * v.x);
    atomicAdd(out + 1, w * v.y);
    atomicAdd(out + 2, w * v.z);
    atomicAdd(out + 3, w * v.w);
}

// ---------------- per-column sum / sumsq (blockDim.x == d) ----------------
__global__ void colstats(const float* __restrict__ X, float* __restrict__ colsum,
                         float* __restrict__ colsq, int n, int d) {
    int f = threadIdx.x;
    int rows_per = (n + gridDim.x - 1) / gridDim.x;
    int r0 = blockIdx.x * rows_per;
    int r1 = r0 + rows_per; if (r1 > n) r1 = n;
    float s = 0.0f, sq = 0.0f;
    for (int i = r0; i < r1; ++i) {
        float v = X[(long long)i * d + f];
        s += v; sq += v * v;
    }
    atomicAdd(&colsum[f], s);
    atomicAdd(&colsq[f], sq);
}

// ---------------- fused BatchNorm (batch stats, biased var) + ReLU ----------------
// Bias b of the GCN layer cancels in (x - mean), so it is exactly omitted.
__global__ void bn_relu(const float* __restrict__ X, const float* __restrict__ colsum,
                        const float* __restrict__ colsq, const float* __restrict__ g,
                        const float* __restrict__ beta, float* __restrict__ out,
                        int n, int d) {
    long long t = blockIdx.x * (long long)blockDim.x + threadIdx.x;
    if (t >= (long long)n * d) return;
    int f = (int)(t % d);
    float inv_n = 1.0f / (float)n;
    float mean = colsum[f] * inv_n;
    float var = colsq[f] * inv_n - mean * mean;
    float v = (X[t] - mean) * rsqrtf(var + BN_EPS) * g[f] + beta[f];
    out[t] = v > 0.0f ? v : 0.0f;
}

// ---------------- global mean pool ----------------
__global__ void pool_accum(const float* __restrict__ X, const int* __restrict__ batch,
                           float* __restrict__ psum, float* __restrict__ pcnt, int n, int d) {
    long long t = blockIdx.x * (long long)blockDim.x + threadIdx.x;
    if (t >= (long long)n * d) return;
    int i = (int)(t / d);
    int f = (int)(t - (long long)i * d);
    int gi = batch[i];
    atomicAdd(&psum[gi * d + f], X[t]);
    if (f == 0) atomicAdd(&pcnt[gi], 1.0f);
}

__global__ void pool_div(float* __restrict__ psum, const float* __restrict__ pcnt,
                         int gcount, int d) {
    int t = blockIdx.x * blockDim.x + threadIdx.x;
    if (t >= gcount * d) return;
    float c = pcnt[t / d];
    psum[t] /= (c > 1.0f ? c : 1.0f);
}

// ---------------- tiny FC2 (Nc=10, not WMMA-shaped) ----------------
__global__ void fc2_kernel(const float* __restrict__ Hin, const float* __restrict__ W,
                           const float* __restrict__ b, float* __restrict__ out,
                           int M, int K, int Nc) {
    int t = blockIdx.x * blockDim.x + threadIdx.x;
    if (t >= M * Nc) return;
    int m = t / Nc, c = t - (t / Nc) * Nc;
    float s = b[c];
    for (int k = 0; k < K; ++k) s += Hin[(long long)m * K + k] * W[(long long)k * Nc + c];
    out[t] = s;
}

// =========================================================================================
extern "C" void kernel_launch(void* const* d_in, const int* in_sizes, int n_in,
                              void* d_out, int out_size, void* d_ws, size_t ws_size,
                              hipStream_t stream) {
    const float* x     = (const float*)d_in[0];
    const int*   ei    = (const int*)d_in[1];
    const int*   batch = (const int*)d_in[2];

    const int N  = in_sizes[2];          // 50000 (multiple of 16)
    const int E  = in_sizes[1] / 2;      // 800000
    const int F0 = in_sizes[0] / N;      // 96
    const int D  = 128;                  // hidden width of GCN layers
    const int Gp = 64;                   // graphs
    const int H3 = 256;                  // FC1 width
    const int NC = in_sizes[18];         // 10

    const int* src = ei;
    const int* dst = ei + E;

    // workspace carve-up (floats)
    float* ws   = (float*)d_ws;
    float* deg  = ws;  ws += N;
    float* dinv = ws;  ws += N;
    float* nrm  = ws;  ws += E;
    float* H    = ws;  ws += (long long)N * D;
    float* AGG  = ws;  ws += (long long)N * D;
    float* XC   = ws;  ws += (long long)N * D;
    float* csum = ws;  ws += 256;
    float* csq  = ws;  ws += 256;
    float* psum = ws;  ws += Gp * D;
    float* pcnt = ws;  ws += Gp;
    float* fc1  = ws;  ws += Gp * H3;

    const int T = 256;
    auto blocksLL = [](long long n, int t) { return (unsigned)((n + t - 1) / t); };

    // ---- GCN normalization: deg(+self loop) -> dinv -> edge norms ----
    fill_f32<<<blocksLL(N, T), T, 0, stream>>>(deg, N, 1.0f);   // self-loop contributes 1
    degree_kernel<<<blocksLL(E, T), T, 0, stream>>>(dst, deg, E);
    dinv_kernel<<<blocksLL(N, T), T, 0, stream>>>(deg, dinv, N);
    norm_kernel<<<blocksLL(E, T), T, 0, stream>>>(src, dst, dinv, nrm, E);

    // ---- three GCNConv + BN + ReLU layers ----
    const float* Xin = x;
    int din = F0;
    for (int l = 0; l < 3; ++l) {
        const float* W    = (const float*)d_in[3 + l * 4];
        const float* g    = (const float*)d_in[5 + l * 4];
        const float* beta = (const float*)d_in[6 + l * 4];

        // H = Xin @ W   (FP32 WMMA, one wave per 16x16 tile)
        {
            int tiles = (N / 16) * (D / 16);
            wmma_gemm_f32<<<(tiles + 7) / 8, 256, 0, stream>>>(Xin, W, H, nullptr, N, din, D, 0);
        }
        // AGG = dinv^2 * H  (self loops, full overwrite)
        selfloop_init<<<blocksLL((long long)N * D, T), T, 0, stream>>>(H, dinv, AGG, N, D);
        // AGG[dst] += norm * H[src]
        edge_scatter<<<blocksLL((long long)E * (D / 4), T), T, 0, stream>>>(src, dst, nrm, H, AGG, E, D);
        // batch stats + BN + ReLU  (layer bias cancels inside BN)
        fill_f32<<<1, 256, 0, stream>>>(csum, 256, 0.0f);   // csum & csq are contiguous? zero both:
        fill_f32<<<1, 256, 0, stream>>>(csq, 256, 0.0f);
        colstats<<<256, D, 0, stream>>>(AGG, csum, csq, N, D);
        bn_relu<<<blocksLL((long long)N * D, T), T, 0, stream>>>(AGG, csum, csq, g, beta, XC, N, D);

        Xin = XC;
        din = D;
    }

    // ---- global mean pool ----
    fill_f32<<<blocksLL(Gp * D + Gp, T), T, 0, stream>>>(psum, (long long)Gp * D + Gp, 0.0f); // psum then pcnt (contiguous)
    pool_accum<<<blocksLL((long long)N * D, T), T, 0, stream>>>(XC, batch, psum, pcnt, N, D);
    pool_div<<<blocksLL(Gp * D, T), T, 0, stream>>>(psum, pcnt, Gp, D);

    // ---- FC1: relu(pooled @ Wf1 + bf1) via WMMA ----
    {
        const float* Wf1 = (const float*)d_in[15];
        const float* bf1 = (const float*)d_in[16];
        int tiles = (Gp / 16) * (H3 / 16);
        wmma_gemm_f32<<<(tiles + 7) / 8, 256, 0, stream>>>(psum, Wf1, fc1, bf1, Gp, D, H3, 1);
    }

    // ---- FC2 ----
    {
        const float* Wf2 = (const float*)d_in[17];
        const float* bf2 = (const float*)d_in[18];
        fc2_kernel<<<blocksLL(Gp * NC, T), T, 0, stream>>>(fc1, Wf2, bf2, (float*)d_out, Gp, H3, NC);
    }
}